// ICFM_24859270709747
// MI455X (gfx1250) — compile-verified
//
#include <hip/hip_runtime.h>

// ---------------------------------------------------------------------------
// Factorization-machine style interaction kernel for MI455X (gfx1250).
// out[seg] = sum over interactions t with seg_ids[t]==seg of
//            intr_W[intr_idxs[t]] / intr_divs[t] * dot(vecs[f0], vecs[f1]) + b
//
// Memory-bound gather workload (~560 MB / launch, vecs table fits in 192MB L2).
// Dot products of 16 interactions are batched through V_WMMA_F32_16X16X4_F32
// (compute is free at 0.25 FLOP/byte); diagonal of the 16x16 accumulator
// holds the 16 dots. Sorted segment ids -> wave32 segmented scan so only
// run tails issue global f32 atomics.
// ---------------------------------------------------------------------------

typedef float v2f __attribute__((ext_vector_type(2)));
typedef float v8f __attribute__((ext_vector_type(8)));

#define VEC 64  // feature vector length (floats)

__global__ void ICFM_zero_kernel(float* __restrict__ out, int n) {
    int i = blockIdx.x * blockDim.x + threadIdx.x;
    if (i < n) out[i] = 0.0f;
}

__global__ __launch_bounds__(256) void ICFM_24859270709747_kernel(
    const int*   __restrict__ intr_idxs,   // [T]
    const float* __restrict__ intr_divs,   // [T]
    const int*   __restrict__ feat_idxs,   // [T,2]
    const int*   __restrict__ seg_ids,     // [T] sorted
    const float* __restrict__ vecs,        // [N_FEATS, 64]
    const float* __restrict__ intr_W,      // [N_INTRS]
    const float* __restrict__ intr_b,      // [1]
    float*       __restrict__ out,         // [NUM_SEGMENTS]
    int T)
{
    // 8 waves/block, 8 accumulator VGPRs x 32 lanes per wave for diag bounce
    __shared__ float diag_lds[8 * 8 * 32];

    const int lane = threadIdx.x & 31;
    const int wid  = threadIdx.x >> 5;
    float* wlds = diag_lds + wid * 256;

    const int waveGlobal = blockIdx.x * (blockDim.x >> 5) + wid;
    const int waveCount  = gridDim.x * (blockDim.x >> 5);
    const int nChunks    = T >> 4;   // 16 interactions per wave-chunk (T % 16 == 0)

    const float bias = intr_b[0];
    const int m    = lane & 15;        // interaction slot within chunk
    const int koff = (lane >> 4) << 1; // lanes 16-31 carry K+2 per WMMA layout

    for (int chunk = waveGlobal; chunk < nChunks; chunk += waveCount) {
        const int t0 = chunk << 4;
        const int t  = t0 + m;

        // Row pointers for this lane's WMMA fragment slot.
        const int i0 = feat_idxs[2 * t + 0];
        const int i1 = feat_idxs[2 * t + 1];
        const float* r0 = vecs + (size_t)i0 * VEC + koff;  // A: 16x64 rows
        const float* r1 = vecs + (size_t)i1 * VEC + koff;  // B: 64x16 cols

        // D(16x16) += A(16x4) * B(4x16), chained over K = 64 in steps of 4.
        // Lane L<16 loads (K=4k,4k+1) of row idx[L]; lane L+16 loads (4k+2,4k+3)
        // of the same row -> exact V_WMMA f32 A/B VGPR layout, 16B/row granules.
        v8f acc = {};
        #pragma unroll
        for (int kk = 0; kk < 16; ++kk) {
            v2f a = *reinterpret_cast<const v2f*>(r0 + 4 * kk);
            v2f b = *reinterpret_cast<const v2f*>(r1 + 4 * kk);
            acc = __builtin_amdgcn_wmma_f32_16x16x4_f32(
                /*neg_a=*/false, a, /*neg_b=*/false, b,
                /*c_mod=*/(short)0, acc, /*reuse_a=*/false, /*reuse_b=*/false);
        }

        // Extract diagonal C[m][m] via LDS bounce (same-wave DS ops are
        // in-order; compiler inserts the dscnt wait for the load results).
        // C layout: VGPR r, lanes 0-15 -> M=r,N=lane; lanes 16-31 -> M=r+8,N=lane-16.
        #pragma unroll
        for (int r = 0; r < 8; ++r) wlds[r * 32 + lane] = acc[r];
        float D = (m < 8) ? wlds[m * 32 + m]
                          : wlds[(m - 8) * 32 + (m + 16)];

        // Per-interaction scalar term (lanes 0-15 own real interactions).
        const float W  = intr_W[intr_idxs[t]];
        const float dv = intr_divs[t];
        float val = (lane < 16) ? (W / dv * D + bias) : 0.0f;
        int   seg = (lane < 16) ? seg_ids[t] : -1;

        // Segmented inclusive scan over lanes (seg_ids sorted -> runs).
        #pragma unroll
        for (int off = 1; off <= 8; off <<= 1) {
            float ov = __shfl_up(val, off);
            int   os = __shfl_up(seg, off);
            if (lane >= off && os == seg) val += ov;
        }
        int segNext = __shfl_down(seg, 1);
        bool tail = (lane < 16) && ((lane == 15) || (segNext != seg));
        if (tail) {
            unsafeAtomicAdd(&out[seg], val);  // global_atomic_add_f32
        }
        // Full reconvergence here -> EXEC all-ones before next WMMA burst.
    }
}

extern "C" void kernel_launch(void* const* d_in, const int* in_sizes, int n_in,
                              void* d_out, int out_size, void* d_ws, size_t ws_size,
                              hipStream_t stream) {
    const int*   intr_idxs = (const int*)  d_in[0];
    const float* intr_divs = (const float*)d_in[1];
    const int*   feat_idxs = (const int*)  d_in[2];
    const int*   seg_ids   = (const int*)  d_in[3];
    const float* vecs      = (const float*)d_in[4];
    const float* intr_W    = (const float*)d_in[5];
    const float* intr_b    = (const float*)d_in[6];
    float*       out       = (float*)d_out;
    const int T = in_sizes[0];

    // d_out is poisoned by the harness; zero it every call (atomic-accumulated).
    ICFM_zero_kernel<<<(out_size + 255) / 256, 256, 0, stream>>>(out, out_size);

    // 2048 blocks x 8 waves = 16384 waves; 65536 chunks -> 4 chunks/wave.
    const int blocks = 2048;
    ICFM_24859270709747_kernel<<<blocks, 256, 0, stream>>>(
        intr_idxs, intr_divs, feat_idxs, seg_ids, vecs, intr_W, intr_b, out, T);
}